// Attn_76424648065726
// MI455X (gfx1250) — compile-verified
//
#include <hip/hip_runtime.h>
#include <math.h>

#define H2 4096
#define SEQ 8192

typedef float v2f __attribute__((ext_vector_type(2)));
typedef float v8f __attribute__((ext_vector_type(8)));

// ---------------- Phase 1a: partial[oc][h] = sum_{o in chunk} W[o][h] * w[o] --------
// W is [H2(out) x H2(in)] row-major; we need v[h] = sum_o W[o][h]*w[o] (column sums).
// Consecutive threads take consecutive h -> every global load is a fully coalesced
// 1KB burst. o split into 16 chunks of 256 for parallelism; deterministic 2-stage sum.
__global__ void k_wTw_partial(const float* __restrict__ W,
                              const float* __restrict__ w,
                              float* __restrict__ partial) {
    __shared__ float wsh[256];
    const int tid = threadIdx.x;
    const int h   = blockIdx.x * 256 + tid;
    const int o0  = blockIdx.y * 256;
    wsh[tid] = w[o0 + tid];
    __syncthreads();
    float acc = 0.0f;
    const float* Wp = W + (size_t)o0 * H2 + h;
    #pragma unroll 4
    for (int o = 0; o < 256; ++o)
        acc = fmaf(Wp[(size_t)o * H2], wsh[o], acc);
    partial[blockIdx.y * H2 + h] = acc;
}

// ---------------- Phase 1b: v[h] = sum over 16 partial chunks -----------------------
__global__ void k_vreduce(const float* __restrict__ partial, float* __restrict__ v) {
    const int h = blockIdx.x * 256 + threadIdx.x;
    float acc = 0.0f;
    #pragma unroll
    for (int i = 0; i < 16; ++i) acc += partial[i * H2 + h];
    v[h] = acc;
}

// ---------------- Phase 2: e[s] = dot(outputs[s,:], v) via V_WMMA_F32_16X16X4_F32 ---
// Block = 256 threads (8 waves), owns 16 sequence rows. Each wave takes a 512-wide
// K slice and chains 128 WMMA accumulations:
//   B (4x16): lane n<16 holds {K0,K1} of row s0+n, lane n+16 holds {K2,K3}  (b64 load)
//   A (16x4): v[k..k+3] broadcast to all M rows (from LDS-staged v)
// => D[m][n] = dot-partial of row s0+n, replicated over m; lane n reads C vgpr0.
// Wave partials combined through an LDS tree (deterministic, no atomics).
__global__ void k_energy_wmma(const float* __restrict__ outputs,
                              const float* __restrict__ v,
                              float* __restrict__ e) {
    __shared__ float vs[H2];        // 16 KB
    __shared__ float red[8][16];
    const int tid  = threadIdx.x;
    const int s0   = blockIdx.x * 16;
    for (int i = tid; i < H2; i += 256) vs[i] = v[i];
    __syncthreads();

    const int lane = tid & 31;
    const int wave = tid >> 5;                    // 0..7
    const int row  = s0 + (lane & 15);
    const int koff = (lane >= 16) ? 2 : 0;
    const float* rowp = outputs + (size_t)row * H2;

    v8f c = {0.f, 0.f, 0.f, 0.f, 0.f, 0.f, 0.f, 0.f};
    const int kbeg = wave * 512;
    #pragma unroll 4
    for (int k = kbeg; k < kbeg + 512; k += 4) {
        v2f a, b;
        a.x = vs[k + koff];
        a.y = vs[k + koff + 1];
        b = *(const v2f*)(rowp + k + koff);       // 8B-aligned global_load_b64
        c = __builtin_amdgcn_wmma_f32_16x16x4_f32(
                /*neg_a=*/false, a, /*neg_b=*/false, b,
                /*c_mod=*/(short)0, c, /*reuse_a=*/false, /*reuse_b=*/false);
    }
    if (lane < 16) red[wave][lane] = c[0];        // C vgpr0, lanes 0-15: M=0, N=lane
    __syncthreads();
    if (tid < 16) {
        float acc = 0.0f;
        #pragma unroll
        for (int wv = 0; wv < 8; ++wv) acc += red[wv][tid];
        e[s0 + tid] = acc;
    }
}

// ---------------- Phase 3: softmax over the 8192 energies (single block) ------------
__global__ void k_softmax(const float* __restrict__ e, float* __restrict__ out) {
    __shared__ float red[1024];
    const int t = threadIdx.x;
    float ev[8];
    float mx = -INFINITY;
    #pragma unroll
    for (int i = 0; i < 8; ++i) {
        ev[i] = e[t + i * 1024];
        mx = fmaxf(mx, ev[i]);
    }
    red[t] = mx;
    __syncthreads();
    for (int s = 512; s > 0; s >>= 1) {
        if (t < s) red[t] = fmaxf(red[t], red[t + s]);
        __syncthreads();
    }
    mx = red[0];
    __syncthreads();
    float sum = 0.0f;
    #pragma unroll
    for (int i = 0; i < 8; ++i) {
        ev[i] = __expf(ev[i] - mx);
        sum += ev[i];
    }
    red[t] = sum;
    __syncthreads();
    for (int s = 512; s > 0; s >>= 1) {
        if (t < s) red[t] += red[t + s];
        __syncthreads();
    }
    const float inv = 1.0f / red[0];
    #pragma unroll
    for (int i = 0; i < 8; ++i) out[t + i * 1024] = ev[i] * inv;
}

extern "C" void kernel_launch(void* const* d_in, const int* in_sizes, int n_in,
                              void* d_out, int out_size, void* d_ws, size_t ws_size,
                              hipStream_t stream) {
    (void)in_sizes; (void)n_in; (void)out_size; (void)ws_size;
    const float* outputs = (const float*)d_in[0];   // [SEQ, H2]
    const float* W       = (const float*)d_in[1];   // [H2, H2]
    // d_in[2] = b : unused — softmax is shift-invariant, <b,w> cancels exactly.
    const float* w       = (const float*)d_in[3];   // [H2]
    float* out = (float*)d_out;                     // [1,1,SEQ] -> SEQ floats

    float* ws      = (float*)d_ws;
    float* v       = ws;                            // H2 floats
    float* e       = ws + H2;                       // SEQ floats
    float* partial = ws + H2 + SEQ;                 // 16*H2 floats (~256 KB)

    k_wTw_partial<<<dim3(H2 / 256, 16), 256, 0, stream>>>(W, w, partial);
    k_vreduce    <<<H2 / 256, 256, 0, stream>>>(partial, v);
    k_energy_wmma<<<SEQ / 16, 256, 0, stream>>>(outputs, v, e);
    k_softmax    <<<1, 1024, 0, stream>>>(e, out);
}